// GraphSAGE2_60352880443979
// MI455X (gfx1250) — compile-verified
//
#include <hip/hip_runtime.h>
#include <math.h>

#define NN   100000
#define FIN  128
#define HID  256
#define NE   600000
#define CNT_PAD 102400   // floats, 16B-aligned pad for the count buffer

typedef float v2f __attribute__((ext_vector_type(2)));
typedef float v8f __attribute__((ext_vector_type(8)));

__device__ __forceinline__ float wave_sum32(float v) {
#pragma unroll
    for (int off = 16; off > 0; off >>= 1) v += __shfl_xor(v, off, 32);
    return v;
}

__global__ __launch_bounds__(256) void k_zero(float4* p, long n4) {
    long i = (long)blockIdx.x * blockDim.x + threadIdx.x;
    long stride = (long)gridDim.x * blockDim.x;
    float4 z = make_float4(0.f, 0.f, 0.f, 0.f);
    for (; i < n4; i += stride) p[i] = z;
}

__global__ __launch_bounds__(256) void k_count(const int* __restrict__ dst,
                                               float* __restrict__ cnt) {
    int e = blockIdx.x * blockDim.x + threadIdx.x;
    if (e < NE) unsafeAtomicAdd(&cnt[dst[e]], 1.0f);
}

// One thread per (edge, 4-float chunk): gather feat[src] row, atomic-add into acc[dst] row.
template <int F>
__global__ __launch_bounds__(256) void k_scatter(const int* __restrict__ src,
                                                 const int* __restrict__ dst,
                                                 const float* __restrict__ feat,
                                                 float* __restrict__ acc) {
    const int CH = F / 4;
    long tid = (long)blockIdx.x * blockDim.x + threadIdx.x;
    if (tid >= (long)NE * CH) return;
    int e = (int)(tid / CH);
    int c = (int)(tid % CH) * 4;
    int s = src[e], d = dst[e];
    const float4 v = *(const float4*)(feat + (long)s * F + c);
    float* a = acc + (long)d * F + c;
    unsafeAtomicAdd(a + 0, v.x);
    unsafeAtomicAdd(a + 1, v.y);
    unsafeAtomicAdd(a + 2, v.z);
    unsafeAtomicAdd(a + 3, v.w);
}

// Layer-1: out[m][n] = b1[n] + sum_k (s1[m][k]/max(cnt,1))*Wl[n][k] + sum_k x[m][k]*Wr[n][k]
// One wave per 16-node x 64-col tile, v_wmma_f32_16x16x4_f32, K in steps of 4.
__global__ __launch_bounds__(256) void k_gemm1(const float* __restrict__ s1,
                                               const float* __restrict__ cnt,
                                               const float* __restrict__ x,
                                               const float* __restrict__ Wl,
                                               const float* __restrict__ b1,
                                               const float* __restrict__ Wr,
                                               float* __restrict__ out) {
    int wave = (blockIdx.x * blockDim.x + threadIdx.x) >> 5;
    int lane = threadIdx.x & 31;
    int g  = wave & 3;          // 64-column group (4 groups cover HID=256)
    int r  = wave >> 2;         // 16-row node tile (6250 tiles cover NN)
    int m0 = r * 16;
    int half = lane >> 4;       // 0: K lo pair, 1: K hi pair
    int lp   = lane & 15;
    int nb   = g * 64;
    int row  = m0 + lp;         // A-matrix row this lane supplies
    int koff = half * 2;

    float rc = 1.0f / fmaxf(cnt[row], 1.0f);
    const float* arow_s = s1 + (long)row * FIN;
    const float* arow_x = x  + (long)row * FIN;
    __builtin_prefetch(Wl + (long)(nb + lp) * FIN, 0, 3);
    __builtin_prefetch(Wr + (long)(nb + lp) * FIN, 0, 3);

    v8f acc[4];
#pragma unroll
    for (int t = 0; t < 4; ++t) {
        float bv = b1[nb + t * 16 + lp];   // column bias, same for all 8 rows in acc
#pragma unroll
        for (int j = 0; j < 8; ++j) acc[t][j] = bv;
    }

    // Pass 1: mean-aggregated neighbors x W1_l  (mean folded into A as rc)
#pragma unroll 4
    for (int kb = 0; kb < FIN; kb += 4) {
        v2f a;
        a[0] = arow_s[kb + koff + 0] * rc;
        a[1] = arow_s[kb + koff + 1] * rc;
#pragma unroll
        for (int t = 0; t < 4; ++t) {
            const float* wr = Wl + (long)(nb + t * 16 + lp) * FIN + kb + koff;
            v2f b; b[0] = wr[0]; b[1] = wr[1];
            acc[t] = __builtin_amdgcn_wmma_f32_16x16x4_f32(
                false, a, false, b, (short)0, acc[t], false, false);
        }
    }
    // Pass 2: self features x W1_r
#pragma unroll 4
    for (int kb = 0; kb < FIN; kb += 4) {
        v2f a;
        a[0] = arow_x[kb + koff + 0];
        a[1] = arow_x[kb + koff + 1];
#pragma unroll
        for (int t = 0; t < 4; ++t) {
            const float* wr = Wr + (long)(nb + t * 16 + lp) * FIN + kb + koff;
            v2f b; b[0] = wr[0]; b[1] = wr[1];
            acc[t] = __builtin_amdgcn_wmma_f32_16x16x4_f32(
                false, a, false, b, (short)0, acc[t], false, false);
        }
    }

    // C/D layout: lane L vgpr j -> (m = m0 + 8*(L/16) + j, n = nb + 16*t + L%16)
#pragma unroll
    for (int t = 0; t < 4; ++t) {
        int n = nb + t * 16 + lp;
#pragma unroll
        for (int j = 0; j < 8; ++j) {
            int m = m0 + half * 8 + j;
            out[(long)m * HID + n] = acc[t][j];
        }
    }
}

// In-place L2-normalize + ReLU over 256 features; one wave per node.
__global__ __launch_bounds__(256) void k_norm_relu(float* __restrict__ h) {
    int node = (blockIdx.x * blockDim.x + threadIdx.x) >> 5;
    int lane = threadIdx.x & 31;
    float* row = h + (long)node * HID;
    float v[8];
    float ss = 0.f;
#pragma unroll
    for (int i = 0; i < 8; ++i) { v[i] = row[lane + 32 * i]; ss += v[i] * v[i]; }
    ss = wave_sum32(ss);
    float sc = 1.0f / fmaxf(sqrtf(ss), 1e-12f);
#pragma unroll
    for (int i = 0; i < 8; ++i) row[lane + 32 * i] = fmaxf(v[i] * sc, 0.0f);
}

// Layer-2 (K=256 -> N=2) fused with L2-normalize + log_softmax; one wave per node.
__global__ __launch_bounds__(256) void k_layer2(const float* __restrict__ s2,
                                                const float* __restrict__ cnt,
                                                const float* __restrict__ h,
                                                const float* __restrict__ Wl,
                                                const float* __restrict__ b2,
                                                const float* __restrict__ Wr,
                                                float* __restrict__ out) {
    int node = (blockIdx.x * blockDim.x + threadIdx.x) >> 5;
    int lane = threadIdx.x & 31;
    float rc = 1.0f / fmaxf(cnt[node], 1.0f);
    const float* srow = s2 + (long)node * HID;
    const float* hrow = h  + (long)node * HID;
    float a0 = 0.f, a1 = 0.f, r0 = 0.f, r1 = 0.f;
#pragma unroll
    for (int i = 0; i < 8; ++i) {
        int k = lane + 32 * i;
        float ag = srow[k] * rc;
        float hv = hrow[k];
        a0 += ag * Wl[k];       a1 += ag * Wl[HID + k];
        r0 += hv * Wr[k];       r1 += hv * Wr[HID + k];
    }
    a0 = wave_sum32(a0); a1 = wave_sum32(a1);
    r0 = wave_sum32(r0); r1 = wave_sum32(r1);
    if (lane == 0) {
        float o0 = b2[0] + a0 + r0;
        float o1 = b2[1] + a1 + r1;
        float sc = 1.0f / fmaxf(sqrtf(o0 * o0 + o1 * o1), 1e-12f);
        o0 *= sc; o1 *= sc;
        float mx  = fmaxf(o0, o1);
        float lse = mx + logf(expf(o0 - mx) + expf(o1 - mx));
        out[(long)node * 2 + 0] = o0 - lse;
        out[(long)node * 2 + 1] = o1 - lse;
    }
}

extern "C" void kernel_launch(void* const* d_in, const int* in_sizes, int n_in,
                              void* d_out, int out_size, void* d_ws, size_t ws_size,
                              hipStream_t stream) {
    const float* x   = (const float*)d_in[0];
    const int*   ei  = (const int*)d_in[1];   // int32 (JAX default demotes int64)
    const float* W1l = (const float*)d_in[2];
    const float* b1  = (const float*)d_in[3];
    const float* W1r = (const float*)d_in[4];
    const float* W2l = (const float*)d_in[5];
    const float* b2  = (const float*)d_in[6];
    const float* W2r = (const float*)d_in[7];
    const int* src = ei;
    const int* dst = ei + NE;

    float* cnt  = (float*)d_ws;                 // NN counts (CNT_PAD floats)
    float* bufA = cnt + CNT_PAD;                // s1 (NN*128) then s2 (NN*256)
    float* hbuf = bufA + (long)NN * HID;        // raw out1, then normalized h (NN*256)
    float* out  = (float*)d_out;

    // zero cnt + s1 region (contiguous)
    {
        long n4 = (CNT_PAD + (long)NN * FIN) / 4;
        k_zero<<<2048, 256, 0, stream>>>((float4*)cnt, n4);
    }
    k_count<<<(NE + 255) / 256, 256, 0, stream>>>(dst, cnt);
    {
        long threads = (long)NE * (FIN / 4);
        k_scatter<FIN><<<(int)((threads + 255) / 256), 256, 0, stream>>>(src, dst, x, bufA);
    }
    // 6250 row-tiles * 4 col-groups = 25000 waves -> 3125 full blocks
    k_gemm1<<<3125, 256, 0, stream>>>(bufA, cnt, x, W1l, b1, W1r, hbuf);
    k_norm_relu<<<(NN * 32) / 256, 256, 0, stream>>>(hbuf);
    // zero s2 region
    {
        long n4 = ((long)NN * HID) / 4;
        k_zero<<<2048, 256, 0, stream>>>((float4*)bufA, n4);
    }
    {
        long threads = (long)NE * (HID / 4);
        k_scatter<HID><<<(int)((threads + 255) / 256), 256, 0, stream>>>(src, dst, hbuf, bufA);
    }
    k_layer2<<<(NN * 32) / 256, 256, 0, stream>>>(bufA, cnt, hbuf, W2l, b2, W2r, out);
}